// DistanceAwareLabelSmoothing_78237124264173
// MI455X (gfx1250) — compile-verified
//
#include <hip/hip_runtime.h>
#include <stdint.h>

// -------------------------------------------------------------------------
// DistanceAwareLabelSmoothing loss for MI455X (gfx1250).
// loss_b = logsumexp(x_b) - [0.9*x_t + (0.1/total[t]) * sum_{c!=t} x_c/(|c-t|+1)]
// Memory-bound: 65.5 MB read once -> ~2.8us floor @ 23.3 TB/s.
// CDNA5 paths used: tensor_load_to_lds (TDM DMA, double-buffered) +
// s_wait_tensorcnt; wave32 shfl_xor butterflies; trans-pipe exp/log.
// -------------------------------------------------------------------------

#define NCLS        1000
#define BATCH       16384
#define SMOOTHING   0.1f
#define CONFIDENCE  0.9f

#define TILE_ROWS       4                 // rows per TDM tile (one per wave)
#define WAVES           4
#define TPB             (WAVES * 32)      // 128 threads = 4 wave32
#define TILES_PER_BLOCK 4
#define NBLK            (BATCH / (TILE_ROWS * TILES_PER_BLOCK))   // 1024

typedef uint32_t u32;
typedef __attribute__((ext_vector_type(4))) u32 u32x4_t;
typedef __attribute__((ext_vector_type(8))) int i32x8_t;
typedef __attribute__((ext_vector_type(4))) int i32x4_t;

// Low 32 bits of a flat pointer into LDS == byte offset within the
// workgroup's LDS allocation (ISA 10.2 aperture mapping).
__device__ __forceinline__ u32 lds_off(const void* p) {
  return (u32)(uintptr_t)p;
}

// Issue one TDM DMA: contiguous 1-D tile of TILE_ROWS*NCLS f32 -> LDS.
// D# layout per CDNA5 ISA.
__device__ __forceinline__ void tdm_load_tile(const float* gsrc, u32 lds_dst) {
  const uint64_t ga = (uint64_t)(uintptr_t)gsrc;
  // group0: count=1 | lds_addr | global_addr[56:0] | type=2 at bits 127:126
  u32x4_t g0 = { 1u,
                 lds_dst,
                 (u32)ga,
                 (u32)((ga >> 32) & 0x1FFFFFFu) | (2u << 30) };
  const int ELEMS = TILE_ROWS * NCLS;   // 4000 elements of 4 bytes
  // group1: data_size=2 (4B); tensor_dim0=tile_dim0=ELEMS; tensor_dim1=tile_dim1=1;
  //         tensor_dim0_stride=ELEMS; no padding, no multicast, no iterate.
  i32x8_t g1 = { 0x20000,          // data_size=2 at bits 17:16
                 ELEMS << 16,      // tensor_dim0[15:0] at bits 63:48
                 1 << 16,          // tensor_dim0[31:16]=0 | tensor_dim1[15:0]=1
                 ELEMS << 16,      // tensor_dim1[31:16]=0 | tile_dim0=ELEMS
                 1,                // tile_dim1=1 | tile_dim2=0
                 ELEMS,            // tensor_dim0_stride[31:0]
                 0, 0 };
  i32x4_t z4 = { 0, 0, 0, 0 };
#if __clang_major__ >= 23
  i32x8_t z8 = { 0, 0, 0, 0, 0, 0, 0, 0 };
  __builtin_amdgcn_tensor_load_to_lds(g0, g1, z4, z4, z8, 0);
#else
  __builtin_amdgcn_tensor_load_to_lds(g0, g1, z4, z4, 0);
#endif
}

// ---- Kernel A: inv_total[t] = SMOOTHING / sum_{c!=t} 1/(|c-t|+1) ----------
__global__ __launch_bounds__(256) void dals_inv_total(float* __restrict__ inv_total) {
  const int t = blockIdx.x * blockDim.x + threadIdx.x;
  if (t >= NCLS) return;
  float total = 0.0f;
  for (int d = 1; d <= t; ++d)            total += 1.0f / (float)(d + 1);
  for (int d = 1; d <= NCLS - 1 - t; ++d) total += 1.0f / (float)(d + 1);
  inv_total[t] = SMOOTHING / total;
}

// ---- Kernel B: main, TDM double-buffered, one wave per row ----------------
__global__ __launch_bounds__(TPB) void dals_main(const float* __restrict__ logits,
                                                 const int*   __restrict__ targets,
                                                 const float* __restrict__ inv_total,
                                                 float*       __restrict__ partials) {
  __shared__ alignas(16) float buf[2][TILE_ROWS * NCLS]; // 2 x 16000 B
  __shared__ float recip_s[NCLS];                        // 1/(d+1) table
  __shared__ float wave_loss[WAVES];

  const int tid  = threadIdx.x;
  const int lane = tid & 31;
  const int wv   = tid >> 5;

  // Exact-division reciprocal table (matches reference's 1/(dist+1) in f32).
  for (int i = tid; i < NCLS; i += TPB)
    recip_s[i] = 1.0f / (float)(i + 1);

  const int tile0 = blockIdx.x * TILES_PER_BLOCK;

  // Prologue: two TDM DMAs in flight (issued by wave 0 only).
  if (tid < 32) {
    tdm_load_tile(logits + (size_t)tile0 * TILE_ROWS * NCLS,       lds_off(&buf[0][0]));
    tdm_load_tile(logits + (size_t)(tile0 + 1) * TILE_ROWS * NCLS, lds_off(&buf[1][0]));
  }

  float acc = 0.0f;
  for (int i = 0; i < TILES_PER_BLOCK; ++i) {
    if (tid < 32) {
      if (i + 1 < TILES_PER_BLOCK) __builtin_amdgcn_s_wait_tensorcnt(1); // oldest done
      else                         __builtin_amdgcn_s_wait_tensorcnt(0);
    }
    __syncthreads();   // publish DMA'd tile (+recip on first iter) to all waves

    const int    row = (tile0 + i) * TILE_ROWS + wv;
    const float* r   = &buf[i & 1][wv * NCLS];
    const int    t   = targets[row];

    // Register-cache the row: 8 x float4 per lane (32 VGPRs), single LDS pass.
    float4 v[8];
    float m = -3.4e38f;
#pragma unroll
    for (int k = 0; k < 8; ++k) {
      const int j = lane + 32 * k;
      if (j < NCLS / 4) {
        v[k] = reinterpret_cast<const float4*>(r)[j];
        m = fmaxf(m, fmaxf(fmaxf(v[k].x, v[k].y), fmaxf(v[k].z, v[k].w)));
      }
    }
#pragma unroll
    for (int o = 16; o > 0; o >>= 1)
      m = fmaxf(m, __shfl_xor(m, o, 32));

    float se = 0.0f, wsum = 0.0f, vt = 0.0f;
#pragma unroll
    for (int k = 0; k < 8; ++k) {
      const int j = lane + 32 * k;
      if (j < NCLS / 4) {
        const float a[4] = { v[k].x, v[k].y, v[k].z, v[k].w };
        const int c0 = j * 4;
#pragma unroll
        for (int s = 0; s < 4; ++s) {
          const int   c = c0 + s;
          const float x = a[s];
          se += __expf(x - m);
          const int d = (c > t) ? (c - t) : (t - c);
          wsum += x * recip_s[d];
          if (c == t) vt = x;
        }
      }
    }
#pragma unroll
    for (int o = 16; o > 0; o >>= 1) {
      se   += __shfl_xor(se,   o, 32);
      wsum += __shfl_xor(wsum, o, 32);
      vt   += __shfl_xor(vt,   o, 32);
    }

    const float it  = inv_total[t];
    const float lse = m + __logf(se);
    const float dot = CONFIDENCE * vt + it * (wsum - vt);  // recip[0]==1
    acc += lse - dot;

    __syncthreads();   // buf[i&1] now free for reuse
    if (tid < 32 && i + 2 < TILES_PER_BLOCK)
      tdm_load_tile(logits + (size_t)(tile0 + i + 2) * TILE_ROWS * NCLS,
                    lds_off(&buf[i & 1][0]));
  }

  if (lane == 0) wave_loss[wv] = acc;
  __syncthreads();
  if (tid == 0) {
    float s = 0.0f;
    for (int w = 0; w < WAVES; ++w) s += wave_loss[w];
    partials[blockIdx.x] = s;   // plain store -> deterministic (no f32 atomics)
  }
}

// ---- Kernel C: final deterministic reduction + mean -----------------------
__global__ __launch_bounds__(256) void dals_reduce(const float* __restrict__ partials,
                                                   int n, float* __restrict__ out) {
  __shared__ float sm[256];
  float s = 0.0f;
  for (int i = threadIdx.x; i < n; i += 256) s += partials[i];
  sm[threadIdx.x] = s;
  __syncthreads();
  for (int o = 128; o > 0; o >>= 1) {
    if (threadIdx.x < o) sm[threadIdx.x] += sm[threadIdx.x + o];
    __syncthreads();
  }
  if (threadIdx.x == 0) out[0] = sm[0] * (1.0f / (float)BATCH);
}

// ---------------------------------------------------------------------------
extern "C" void kernel_launch(void* const* d_in, const int* in_sizes, int n_in,
                              void* d_out, int out_size, void* d_ws, size_t ws_size,
                              hipStream_t stream) {
  (void)in_sizes; (void)n_in; (void)out_size; (void)ws_size;
  const float* logits  = (const float*)d_in[0];
  const int*   targets = (const int*)d_in[1];
  float* ws        = (float*)d_ws;
  float* inv_total = ws;          // 1000 floats
  float* partials  = ws + 1024;   // NBLK floats

  dals_inv_total<<<4, 256, 0, stream>>>(inv_total);
  dals_main<<<NBLK, TPB, 0, stream>>>(logits, targets, inv_total, partials);
  dals_reduce<<<1, 256, 0, stream>>>(partials, NBLK, (float*)d_out);
}